// Model_13546326851955
// MI455X (gfx1250) — compile-verified
//
#include <hip/hip_runtime.h>
#include <hip/hip_bf16.h>
#include <math.h>

// ---------------------------------------------------------------------------
// 4-layer GCN (DGL GraphConv, norm='both') for MI455X / gfx1250.
//   h' = D_in^{-1/2} * A^T * (D_out^{-1/2} h) @ W + b ; tanh on layers 0..2
// Graph side: wave-per-edge scatter-add; both 44MB feature buffers live in
// the 192MB L2, so atomics resolve on-chip. GEMM side: V_WMMA_F32_16X16X4_F32
// with fully padded operands (K->mult of 4, N->mult of 16, W transposed in
// workspace) so the K-loop is pure b64 loads + wmma, no per-lane predication.
// ---------------------------------------------------------------------------

typedef __attribute__((ext_vector_type(2))) float v2f;
typedef __attribute__((ext_vector_type(8))) float v8f;

#define GCN_NODES 50000
#define GCN_EDGES 800000

__global__ void gcn_fill_zero(float* __restrict__ p, int n) {
    int i = blockIdx.x * blockDim.x + threadIdx.x;
    if (i < n) p[i] = 0.0f;
}

__global__ void gcn_degree(const int* __restrict__ src, const int* __restrict__ dst,
                           float* __restrict__ deg_out, float* __restrict__ deg_in, int E) {
    int e = blockIdx.x * blockDim.x + threadIdx.x;
    if (e < E) {
        atomicAdd(&deg_out[src[e]], 1.0f);
        atomicAdd(&deg_in[dst[e]], 1.0f);
    }
}

__global__ void gcn_inv_sqrt(float* __restrict__ d, int n) {
    int i = blockIdx.x * blockDim.x + threadIdx.x;
    if (i < n) d[i] = rsqrtf(fmaxf(d[i], 1.0f));
}

// Zero-padded transposed weight copy: WT[n*KP + k] = W[k*N + n] (else 0).
__global__ void gcn_pad_wt(const float* __restrict__ W, float* __restrict__ WT,
                           int K, int N, int KP, int NP) {
    int i = blockIdx.x * blockDim.x + threadIdx.x;
    if (i >= KP * NP) return;
    int n = i / KP;
    int k = i - n * KP;
    WT[i] = (k < K && n < N) ? W[(size_t)k * N + n] : 0.0f;
}

__global__ void gcn_pad_bias(const float* __restrict__ b, float* __restrict__ bp,
                             int N, int NP) {
    int i = blockIdx.x * blockDim.x + threadIdx.x;
    if (i < NP) bp[i] = (i < N) ? b[i] : 0.0f;
}

// One wave32 per edge: agg[dst] += h[src] * inv_sqrt_out[src]
__global__ void gcn_scatter(const float* __restrict__ h, const float* __restrict__ inv_out,
                            const int* __restrict__ src, const int* __restrict__ dst,
                            float* __restrict__ agg, int F, int Hst, int Ast, int E) {
    int wid  = (blockIdx.x * blockDim.x + threadIdx.x) >> 5;
    int lane = threadIdx.x & 31;
    if (wid >= E) return;                       // wave-uniform branch
    int s = src[wid];
    int d = dst[wid];
    float sc = inv_out[s];
    const float* hs = h + (size_t)s * Hst;
    float*       ad = agg + (size_t)d * Ast;
    for (int f = lane; f < F; f += 32)
        atomicAdd(&ad[f], hs[f] * sc);
}

// One wave per (16 x 16*NT) output strip. All operands padded: no per-lane
// branching anywhere in the K-loop, EXEC stays all-1s for every WMMA.
//   A  : [M x KP] row-major (agg, pad cols = 0), scaled by inv_in[row]
//   WT : [NP x KP] (transposed, zero-padded weights)
//   out: [M x Nst], cols >= Ncols not written (only last layer differs)
// V_WMMA_F32_16X16X4_F32 layout (wave32):
//   A 16x4 : lanes 0-15 -> M=lane, {v0,v1}={K0,K1}; lanes 16-31 -> {K2,K3}
//   B 4x16 : lanes 0-15 -> N=lane, {v0,v1}={K0,K1}; lanes 16-31 -> {K2,K3}
//   C/D    : 8 VGPRs, vr -> M = r + 8*(lane>=16), N = lane%16
template<int KP, int NT>
__global__ void gcn_gemm_wmma(const float* __restrict__ A, const float* __restrict__ inv_in,
                              const float* __restrict__ WT, const float* __restrict__ biasp,
                              float* __restrict__ out, int tilesM, int groups,
                              int Nst, int Ncols, int do_tanh) {
    int gtid = blockIdx.x * blockDim.x + threadIdx.x;
    int wid  = gtid >> 5;
    int lane = threadIdx.x & 31;
    if (wid >= tilesM * groups) return;         // wave-uniform: EXEC all-1s below
    int tm = wid / groups;
    int tg = wid - tm * groups;
    int m0 = tm << 4;
    int n0 = tg * (16 * NT);

    int lh = lane & 15;     // M for A-operand, N for B-operand
    int kh = lane >> 4;     // 0 -> K pair {0,1}, 1 -> K pair {2,3}

    int row = m0 + lh;                           // M = 50000 = 3125*16: always valid
    float s = inv_in[row];
    const v2f* ap = (const v2f*)(A + (size_t)row * KP + 2 * kh);

    int col0 = n0 + lh;
    const v2f* bp[NT];
#pragma unroll
    for (int t = 0; t < NT; ++t)
        bp[t] = (const v2f*)(WT + (size_t)(col0 + 16 * t) * KP + 2 * kh);

    v8f acc[NT];
#pragma unroll
    for (int t = 0; t < NT; ++t) acc[t] = (v8f){};

#pragma unroll 5
    for (int k0 = 0; k0 < KP; k0 += 4) {
        v2f av = ap[k0 >> 1];
        av.x *= s;
        av.y *= s;
#pragma unroll
        for (int t = 0; t < NT; ++t) {
            v2f bv = bp[t][k0 >> 1];
            acc[t] = __builtin_amdgcn_wmma_f32_16x16x4_f32(
                false, av, false, bv, (short)0, acc[t], false, false);
        }
    }

    int mbase = m0 + 8 * kh;
#pragma unroll
    for (int t = 0; t < NT; ++t) {
        int col = col0 + 16 * t;
        if (col < Ncols) {                       // post-WMMA guard only
            float bb = biasp[col];
#pragma unroll
            for (int r = 0; r < 8; ++r) {
                float v = acc[t][r] + bb;
                out[(size_t)(mbase + r) * Nst + col] = do_tanh ? tanhf(v) : v;
            }
        }
    }
}

static inline int gcn_cdiv(int a, int b) { return (a + b - 1) / b; }

extern "C" void kernel_launch(void* const* d_in, const int* in_sizes, int n_in,
                              void* d_out, int out_size, void* d_ws, size_t ws_size,
                              hipStream_t stream) {
    (void)in_sizes; (void)n_in; (void)out_size; (void)ws_size;

    const float* x  = (const float*)d_in[0];
    const float* Wl[4] = { (const float*)d_in[1], (const float*)d_in[3],
                           (const float*)d_in[5], (const float*)d_in[7] };
    const float* bl[4] = { (const float*)d_in[2], (const float*)d_in[4],
                           (const float*)d_in[6], (const float*)d_in[8] };
    const int* ei  = (const int*)d_in[9];     // edge_index (int32 in practice)
    const int* src = ei;
    const int* dst = ei + GCN_EDGES;
    float* out = (float*)d_out;

    // Workspace layout (floats)
    float* ws      = (float*)d_ws;
    float* inv_out = ws;                                   // [N] deg_out -> rsqrt
    float* inv_in  = ws + GCN_NODES;                       // [N] deg_in  -> rsqrt
    float* bufA    = ws + 2 * (size_t)GCN_NODES;           // [N x 220] aggregation
    float* bufB    = bufA + (size_t)GCN_NODES * 220;       // [N x 224] hidden state
    float* WT      = bufB + (size_t)GCN_NODES * 224;       // [224 x 224] padded W^T
    float* biasp   = WT + (size_t)224 * 224;               // [224] padded bias

    const int B = 256;
    const int tilesM = GCN_NODES / 16;                     // 3125, exact

    // Degrees -> inv-sqrt (contiguous inv_out|inv_in: one pass for both)
    gcn_fill_zero<<<gcn_cdiv(2 * GCN_NODES, B), B, 0, stream>>>(inv_out, 2 * GCN_NODES);
    gcn_degree  <<<gcn_cdiv(GCN_EDGES, B),     B, 0, stream>>>(src, dst, inv_out, inv_in, GCN_EDGES);
    gcn_inv_sqrt<<<gcn_cdiv(2 * GCN_NODES, B), B, 0, stream>>>(inv_out, 2 * GCN_NODES);

    const int Kdim[4] = { 22, 220, 220, 220 };   // logical K
    const int KPad[4] = { 24, 220, 220, 220 };   // padded K (mult of 4)
    const int Ndim[4] = { 220, 220, 220, 10 };   // logical N
    const int NPad[4] = { 224, 224, 224, 16 };   // padded N (mult of 16)

    const float* hcur = x;
    int hstride = 22;

    for (int L = 0; L < 4; ++L) {
        int K = Kdim[L], KP = KPad[L], N = Ndim[L], NP = NPad[L];

        // Padded transposed weights + padded bias
        gcn_pad_wt  <<<gcn_cdiv(KP * NP, B), B, 0, stream>>>(Wl[L], WT, K, N, KP, NP);
        gcn_pad_bias<<<gcn_cdiv(NP, B),      B, 0, stream>>>(bl[L], biasp, N, NP);

        // Zero + scatter-add aggregation (agg stride = KP, pad cols stay 0)
        int aggN = GCN_NODES * KP;
        gcn_fill_zero<<<gcn_cdiv(aggN, B), B, 0, stream>>>(bufA, aggN);
        gcn_scatter<<<gcn_cdiv(GCN_EDGES * 32, B), B, 0, stream>>>(
            hcur, inv_out, src, dst, bufA, K, hstride, KP, GCN_EDGES);

        // GEMM: out = act((agg * inv_in) @ W + b) via WMMA
        if (L == 3) {
            int groups = NP / 16;                 // NT = 1
            int waves  = tilesM * groups;
            gcn_gemm_wmma<220, 1><<<gcn_cdiv(waves * 32, B), B, 0, stream>>>(
                bufA, inv_in, WT, biasp, out, tilesM, groups, N, N, 0);
        } else if (L == 0) {
            int groups = NP / 32;                 // NT = 2
            int waves  = tilesM * groups;
            gcn_gemm_wmma<24, 2><<<gcn_cdiv(waves * 32, B), B, 0, stream>>>(
                bufA, inv_in, WT, biasp, bufB, tilesM, groups, NP, NP, 1);
        } else {
            int groups = NP / 32;                 // NT = 2
            int waves  = tilesM * groups;
            gcn_gemm_wmma<220, 2><<<gcn_cdiv(waves * 32, B), B, 0, stream>>>(
                bufA, inv_in, WT, biasp, bufB, tilesM, groups, NP, NP, 1);
        }

        hcur = bufB;
        hstride = NP;
    }
}